// Qu4_1099511628330
// MI455X (gfx1250) — compile-verified
//
#include <hip/hip_runtime.h>
#include <hip/hip_bf16.h>

typedef __attribute__((ext_vector_type(2))) float v2f;
typedef __attribute__((ext_vector_type(8))) float v8f;

#define TPB 256

// row-major strictly-upper-triangular index for an 8x8 matrix, r < c
__device__ __forceinline__ int triu_idx(int r, int c) {
  return r * 7 - (r * (r - 1)) / 2 + (c - r - 1);
}

// OBJ[q,w,e,r] for V = 0.1 (computed inline, matches _obj_np)
__device__ __forceinline__ float obj_val(int q, int w, int e, int r) {
  const float V = 0.1f;
  if (q != w && w != e && e != r && q != r) {
    if (q == e || w == r) {
      if (q == e && w == r) return (1.0f - V) * (1.0f / 256.0f);
      return (3.0f + 5.0f * V) * (1.0f / 768.0f);
    }
    return (3.0f + 13.0f * V) * (1.0f / 768.0f);
  }
  return (1.0f - V) * (1.0f / 256.0f);
}

// Single-workgroup kernel: 256 threads = 8 wave32 on one WGP.
// ws needs 16*4096 floats (256 KB) for the 16 B_ab matrices.
__launch_bounds__(TPB)
__global__ void qu4_povm_kernel(const float* __restrict__ prm,
                                float* __restrict__ out,
                                float* __restrict__ ws) {
  const int tid = threadIdx.x;

  __shared__ float psi[8];
  __shared__ float eig[4][8];
  __shared__ float Bsc[3][64];      // scaled antisymmetric generators A/16
  __shared__ float Tbuf[2][3][64];  // ping-pong for expm matmuls
  __shared__ float Mm[4][64];       // POVM elements M0..M3
  __shared__ float S[4096];         // swapped state as 8^4 tensor (i,j,k,l)
  __shared__ float Aq[4096];        // one-leg contraction staging
  __shared__ float partial[8][256]; // per-wave WMMA accumulators
  __shared__ float resS[256];
  __shared__ float red[256];

  // ---- phase 0: psi, softmax eigenvalues, generators -----------------------
  if (tid < 8) {
    // softmax over the 4 rows of prm[5:37].reshape(4,8), per column `tid`
    float e0 = __expf(0.0f); // placeholder to keep fast-math out; use expf below
    (void)e0;
    float x0 = expf(prm[5 + 0 * 8 + tid]);
    float x1 = expf(prm[5 + 1 * 8 + tid]);
    float x2 = expf(prm[5 + 2 * 8 + tid]);
    float x3 = expf(prm[5 + 3 * 8 + tid]);
    float s = x0 + x1 + x2 + x3;
    eig[0][tid] = x0 / s; eig[1][tid] = x1 / s;
    eig[2][tid] = x2 / s; eig[3][tid] = x3 / s;
  }
  if (tid == 8) {
    float a0 = expf(prm[0]), a1 = expf(prm[1]), a2 = expf(prm[2]),
          a3 = expf(prm[3]), a4 = expf(prm[4]);
    float n = sqrtf(a0 * a0 + a1 * a1 + a2 * a2 + a3 * a3 + a4 * a4);
    psi[0] = a0 / n; psi[1] = 0.0f; psi[2] = 0.0f; psi[3] = 0.0f;
    psi[4] = a1 / n; psi[5] = a2 / n; psi[6] = a3 / n; psi[7] = a4 / n;
  }
  if (tid < 192) {
    int a = tid >> 6, e = tid & 63, r = e >> 3, c = e & 7;
    float v = 0.0f;
    if (r < c)      v =  prm[37 + a * 28 + triu_idx(r, c)];
    else if (r > c) v = -prm[37 + a * 28 + triu_idx(c, r)];
    Bsc[a][e] = v * 0.0625f;                 // A / 16  (scaling for squaring)
    Tbuf[0][a][e] = (r == c) ? 1.0f : 0.0f;  // Horner start: T = I
  }
  __syncthreads();

  // ---- phase 1: expm via order-8 Horner Taylor + 4 squarings ---------------
  int cur = 0;
  for (int k = 8; k >= 1; --k) {            // T <- I + (B/k) * T
    if (tid < 192) {
      int a = tid >> 6, e = tid & 63, r = e >> 3, c = e & 7;
      float acc = 0.0f;
#pragma unroll
      for (int t = 0; t < 8; ++t) acc += Bsc[a][r * 8 + t] * Tbuf[cur][a][t * 8 + c];
      Tbuf[1 - cur][a][e] = ((r == c) ? 1.0f : 0.0f) + acc / (float)k;
    }
    __syncthreads();
    cur ^= 1;
  }
  for (int sq = 0; sq < 4; ++sq) {          // E <- E*E, four times  (E = exp(A/16)^16)
    if (tid < 192) {
      int a = tid >> 6, e = tid & 63, r = e >> 3, c = e & 7;
      float acc = 0.0f;
#pragma unroll
      for (int t = 0; t < 8; ++t) acc += Tbuf[cur][a][r * 8 + t] * Tbuf[cur][a][t * 8 + c];
      Tbuf[1 - cur][a][e] = acc;
    }
    __syncthreads();
    cur ^= 1;
  }

  // ---- phase 2: M_a = U diag(e_a) U^T ;  M3 = I - M0 - M1 - M2 -------------
  if (tid < 192) {
    int a = tid >> 6, e = tid & 63, r = e >> 3, c = e & 7;
    float acc = 0.0f;
#pragma unroll
    for (int t = 0; t < 8; ++t)
      acc += Tbuf[cur][a][r * 8 + t] * eig[a][t] * Tbuf[cur][a][c * 8 + t];
    Mm[a][e] = acc;   // symmetric by construction (bitwise)
  }
  __syncthreads();
  if (tid < 64) {
    int r = tid >> 3, c = tid & 7;
    Mm[3][tid] = ((r == c) ? 1.0f : 0.0f) - Mm[0][tid] - Mm[1][tid] - Mm[2][tid];
  }
  __syncthreads();
  out[257 + tid] = Mm[tid >> 6][tid & 63];  // m output [4,8,8]

  // ---- phase 3: swapped state, SWAP folded into a 12-bit permutation -------
  // p = [11,1,3,2,4,6,5,7,9,8,10,0]: state2[(i,j,k,l)] = psi[g0]psi[g1]psi[g2]psi[g3]
  for (int t = 0; t < 16; ++t) {
    int idx = tid + t * 256;
    int i = idx >> 9, j = (idx >> 6) & 7, k = (idx >> 3) & 7, l = idx & 7;
    int g0 = ((l & 1) << 2) | (((i >> 1) & 1) << 1) | ((j >> 2) & 1);
    int g1 = ((i & 1) << 2) | (((j >> 1) & 1) << 1) | ((k >> 2) & 1);
    int g2 = ((j & 1) << 2) | (((k >> 1) & 1) << 1) | ((l >> 2) & 1);
    int g3 = ((k & 1) << 2) | (((l >> 1) & 1) << 1) | ((i >> 2) & 1);
    S[idx] = psi[g0] * psi[g1] * psi[g2] * psi[g3];
  }
  __syncthreads();

  // ---- phase 4: B_qw[(m,n),(k,l)] = sum_{i,j} Mq[m,i] Mw[n,j] S[i,j,k,l] ---
  for (int q = 0; q < 4; ++q) {
    for (int t = 0; t < 16; ++t) {          // Aq[m,j,kl] = sum_i Mq[m,i] S[i,j,kl]
      int idx = tid + t * 256;
      int m = idx >> 9, rem = idx & 511;
      float acc = 0.0f;
#pragma unroll
      for (int i = 0; i < 8; ++i) acc += Mm[q][m * 8 + i] * S[i * 512 + rem];
      Aq[idx] = acc;
    }
    __syncthreads();
    for (int w = 0; w < 4; ++w) {
      float* Bout = ws + (q * 4 + w) * 4096;
      for (int t = 0; t < 16; ++t) {
        int idx = tid + t * 256;
        int m = idx >> 9, n = (idx >> 6) & 7, kl = idx & 63;
        float acc = 0.0f;
#pragma unroll
        for (int j = 0; j < 8; ++j) acc += Mm[w][n * 8 + j] * Aq[m * 512 + j * 64 + kl];
        Bout[idx] = acc;
      }
    }
    __syncthreads();
  }
  __threadfence();   // make ws stores visible before WMMA phase reads them
  __syncthreads();

  // ---- phase 5: result[a,b] = sum_{X,Y} B_a[X,Y] * B_b[Y,X] ----------------
  // 16x16 output, K = 4096, exact f32 via V_WMMA_F32_16X16X4_F32.
  // K split across the 8 waves; each wave does 128 chained WMMAs.
  {
    int lane = tid & 31, wv = tid >> 5;
    int al = lane & 15, hi = lane >> 4;
    v8f acc = {0.f, 0.f, 0.f, 0.f, 0.f, 0.f, 0.f, 0.f};
    int tbeg = wv * 512;
#pragma unroll 4
    for (int t0 = tbeg; t0 < tbeg + 512; t0 += 4) {
      int t = t0 + 2 * hi;                  // this half-wave's K pair
      v2f av, bv;
      // A fragment: P[alpha=M, t] = B_alpha flat
      av.x = ws[al * 4096 + t];
      av.y = ws[al * 4096 + t + 1];
      // B fragment: Q[beta=N, t] = B_beta transposed: t=(X<<6)|Y -> (Y<<6)|X
      int u = t + 1;
      int tp0 = ((t & 63) << 6) | (t >> 6);
      int tp1 = ((u & 63) << 6) | (u >> 6);
      bv.x = ws[al * 4096 + tp0];
      bv.y = ws[al * 4096 + tp1];
      acc = __builtin_amdgcn_wmma_f32_16x16x4_f32(
          /*neg_a=*/false, av, /*neg_b=*/false, bv,
          /*c_mod=*/(short)0, acc, /*reuse_a=*/false, /*reuse_b=*/false);
    }
#pragma unroll
    for (int v = 0; v < 8; ++v)             // D layout: lane<16 -> M=v, else M=v+8
      partial[wv][(v + 8 * hi) * 16 + al] = acc[v];
  }
  __syncthreads();

  // reduce the 8 per-wave partials; emit result[4,4,4,4] (flat = alpha*16+beta)
  {
    float r8 = 0.0f;
#pragma unroll
    for (int wv = 0; wv < 8; ++wv) r8 += partial[wv][tid];
    resS[tid] = r8;
    out[1 + tid] = r8;
  }
  __syncthreads();

  // ---- phase 6: KL divergence ---------------------------------------------
  {
    int q = tid >> 6, w = (tid >> 4) & 3, e = (tid >> 2) & 3, r = tid & 3;
    float o = obj_val(q, w, e, r);
    red[tid] = o * logf(o / resS[tid]);
  }
  __syncthreads();
  if (tid == 0) {
    float s = 0.0f;
    for (int t = 0; t < 256; ++t) s += red[t];
    out[0] = s;
  }
}

extern "C" void kernel_launch(void* const* d_in, const int* in_sizes, int n_in,
                              void* d_out, int out_size, void* d_ws, size_t ws_size,
                              hipStream_t stream) {
  (void)in_sizes; (void)n_in; (void)out_size; (void)ws_size; // ws needs >= 256 KB
  const float* prm = (const float*)d_in[0];   // params[121], f32
  float* out = (float*)d_out;                 // [y(1) | result(256) | m(256)], f32
  float* ws = (float*)d_ws;                   // 16 * 4096 f32 B-matrices
  qu4_povm_kernel<<<1, TPB, 0, stream>>>(prm, out, ws);
}